// ECELoss_32195074850950
// MI455X (gfx1250) — compile-verified
//
#include <hip/hip_runtime.h>
#include <limits.h>

#define N_BINS 15
#define ROW_C 128          // classes per row (reference fixes C=128)
#define BLOCK_THREADS 256  // 8 wave32 per block
#define GRID_BLOCKS 4096
#define ROWS_PER_ITER 4    // rows processed per wave per loop iteration (MLP)

// d_ws layout: [0..14]=cnt, [15..29]=sum_conf, [30..44]=sum_acc  (floats)

__global__ void ece_init_kernel(float* __restrict__ bins) {
    int t = threadIdx.x;
    if (t < 3 * N_BINS) bins[t] = 0.0f;
}

__global__ __launch_bounds__(BLOCK_THREADS)
void ece_main_kernel(const float* __restrict__ logits,
                     const int* __restrict__ labels,
                     float* __restrict__ bins,
                     int n_rows) {
    __shared__ float s_cnt[N_BINS];
    __shared__ float s_conf[N_BINS];
    __shared__ float s_acc[N_BINS];

    const int tid = threadIdx.x;
    if (tid < N_BINS) {
        s_cnt[tid]  = 0.0f;
        s_conf[tid] = 0.0f;
        s_acc[tid]  = 0.0f;
    }
    __syncthreads();

    const int lane          = tid & 31;          // wave32 lane
    const int wave_in_block = tid >> 5;
    const int waves_per_blk = BLOCK_THREADS >> 5;
    const int gwave         = blockIdx.x * waves_per_blk + wave_in_block;
    const int n_waves       = gridDim.x * waves_per_blk;
    const int batch_stride  = n_waves * ROWS_PER_ITER;
    const int base_idx      = lane << 2;         // this lane's first column

    for (int row0 = gwave * ROWS_PER_ITER; row0 < n_rows; row0 += batch_stride) {
        // ---- Issue up to 4 independent global_load_b128 back-to-back (MLP) ----
        float4 v[ROWS_PER_ITER];
        #pragma unroll
        for (int k = 0; k < ROWS_PER_ITER; ++k) {
            int row = row0 + k;                  // wave-uniform guard (scalar branch)
            if (row < n_rows) {
                v[k] = ((const float4*)(logits + (size_t)row * ROW_C))[lane];
            }
        }

        // ---- One global_prefetch_b8 covers the whole next 2KB batch ----
        int pf_row0 = row0 + batch_stride;
        if (pf_row0 + ROWS_PER_ITER <= n_rows) {
            const char* pf = (const char*)(logits + (size_t)pf_row0 * ROW_C);
            __builtin_prefetch(pf + lane * 64, 0, 3);   // 32 lanes * 64B = 2KB
        }

        #pragma unroll
        for (int k = 0; k < ROWS_PER_ITER; ++k) {
            int row = row0 + k;
            if (row >= n_rows) break;            // wave-uniform
            float4 x = v[k];

            // In-lane max (value only)
            float m = x.x;
            if (x.y > m) m = x.y;
            if (x.z > m) m = x.z;
            if (x.w > m) m = x.w;

            // Wave32 butterfly max: 5 shuffles, all lanes end with identical row max
            #pragma unroll
            for (int off = 16; off > 0; off >>= 1) {
                float om = __shfl_xor(m, off, 32);
                if (om > m) m = om;
            }

            // Argmax via ballot: earliest matching element (matches jnp.argmax ties)
            int cand = INT_MAX;
            if (x.w == m) cand = base_idx + 3;   // assign in reverse: earliest wins
            if (x.z == m) cand = base_idx + 2;
            if (x.y == m) cand = base_idx + 1;
            if (x.x == m) cand = base_idx;
            unsigned long long msk = __ballot(cand != INT_MAX);
            int first_lane = (int)__ffsll(msk) - 1;
            int mi = __shfl(cand, first_lane, 32);   // 1 broadcast vs 5-step butterfly

            // conf = max(softmax) = 1 / sum(exp(x - max)); 5-shuffle butterfly sum
            float s = __expf(x.x - m) + __expf(x.y - m) +
                      __expf(x.z - m) + __expf(x.w - m);
            #pragma unroll
            for (int off = 16; off > 0; off >>= 1) {
                s += __shfl_xor(s, off, 32);
            }

            if (lane == 0) {
                float conf = 1.0f / s;
                float acc  = (mi == labels[row]) ? 1.0f : 0.0f;
                // first i with conf <= (i+1)/15  ==  ceil(conf*15)-1, clamped
                int bin = (int)ceilf(conf * (float)N_BINS) - 1;
                bin = bin < 0 ? 0 : (bin > N_BINS - 1 ? N_BINS - 1 : bin);
                atomicAdd(&s_cnt[bin],  1.0f);   // ds_add_f32
                atomicAdd(&s_conf[bin], conf);
                atomicAdd(&s_acc[bin],  acc);
            }
        }
    }

    __syncthreads();
    if (tid < N_BINS) {
        // one global_atomic_add_f32 per bin per block
        atomicAdd(&bins[tid],              s_cnt[tid]);
        atomicAdd(&bins[tid + N_BINS],     s_conf[tid]);
        atomicAdd(&bins[tid + 2 * N_BINS], s_acc[tid]);
    }
}

__global__ void ece_final_kernel(const float* __restrict__ bins,
                                 float* __restrict__ out,
                                 int n_rows) {
    if (threadIdx.x == 0 && blockIdx.x == 0) {
        float inv_n = 1.0f / (float)n_rows;
        float ece = 0.0f;
        #pragma unroll
        for (int i = 0; i < N_BINS; ++i) {
            float c = bins[i];
            if (c > 0.0f) {
                float avg_conf = bins[i + N_BINS]     / c;
                float avg_acc  = bins[i + 2 * N_BINS] / c;
                ece += fabsf(avg_conf - avg_acc) * (c * inv_n);
            }
        }
        out[0] = ece;
    }
}

extern "C" void kernel_launch(void* const* d_in, const int* in_sizes, int n_in,
                              void* d_out, int out_size, void* d_ws, size_t ws_size,
                              hipStream_t stream) {
    const float* logits = (const float*)d_in[0];
    const int*   labels = (const int*)d_in[1];
    float*       out    = (float*)d_out;
    float*       bins   = (float*)d_ws;      // 45 floats of scratch
    const int    n_rows = in_sizes[1];       // labels count = N

    // Zero the bin accumulators every launch (d_ws is poisoned, not re-zeroed).
    ece_init_kernel<<<1, 64, 0, stream>>>(bins);

    const int rows_per_block = (BLOCK_THREADS >> 5) * ROWS_PER_ITER; // 32
    int blocks = (n_rows + rows_per_block - 1) / rows_per_block;
    if (blocks > GRID_BLOCKS) blocks = GRID_BLOCKS;
    if (blocks < 1) blocks = 1;
    ece_main_kernel<<<blocks, BLOCK_THREADS, 0, stream>>>(logits, labels, bins, n_rows);

    ece_final_kernel<<<1, 32, 0, stream>>>(bins, out, n_rows);
}